// Model_15556371546561
// MI455X (gfx1250) — compile-verified
//
#include <hip/hip_runtime.h>
#include <math.h>
#include <stdint.h>

// ---------------------------------------------------------------------------
// CDNA5 (gfx1250) WMMA helpers: V_WMMA_F32_16X16X4_F32, wave32.
// A: 16x4 f32  -> 2 VGPR/lane: lanes 0-15 hold row M=lane, K pair (0,1);
//                 lanes 16-31 hold row M=lane-16, K pair (2,3).
// B: 4x16 f32  -> mirror: lanes hold col N, halves hold K pairs.
// C/D: 16x16   -> 8 VGPR: vgpr v = row v (lanes 0-15) / row v+8 (lanes 16-31).
// ---------------------------------------------------------------------------
typedef float v2f __attribute__((ext_vector_type(2)));
typedef float v8f __attribute__((ext_vector_type(8)));
typedef unsigned int u32x4 __attribute__((ext_vector_type(4)));
typedef int i32x4 __attribute__((ext_vector_type(4)));
typedef int i32x8 __attribute__((ext_vector_type(8)));

__device__ __forceinline__ v8f wmma4(v2f a, v2f b, v8f c) {
  return __builtin_amdgcn_wmma_f32_16x16x4_f32(
      /*neg_a=*/false, a, /*neg_b=*/false, b,
      /*c_mod=*/(short)0, c, /*reuse_a=*/false, /*reuse_b=*/false);
}

// ---------------------------------------------------------------------------
// CDNA5 async data movement.
// Generic pointers into LDS carry the LDS byte offset in their low 32 bits
// (flat aperture mapping, ISA 10.2), which is exactly the VDST operand of
// GLOBAL_LOAD_ASYNC_TO_LDS / the lds_addr field of a TDM descriptor.
// ---------------------------------------------------------------------------
__device__ __forceinline__ void async_b128(const float* lds_dst, const float* gsrc) {
  uint32_t l = (uint32_t)(uintptr_t)lds_dst;
  uint64_t g = (uint64_t)(uintptr_t)gsrc;
  asm volatile("global_load_async_to_lds_b128 %0, %1, off"
               :: "v"(l), "v"(g) : "memory");
}
__device__ __forceinline__ void wait_async0() {
  asm volatile("s_wait_asynccnt 0x0" ::: "memory");
}

// TDM: 1D row of f32 (nelem <= 65535) global -> LDS.  D# per ISA 8.3/8.4:
// g0: count=1 | lds_addr | global_addr[56:0] | type=2
// g1: data_size=4B, tensor_dim0=n, tensor_dim1=1, tile_dim0=n, tile_dim1=1,
//     tensor_dim0_stride=n
__device__ __forceinline__ void tdm_load_row_f32(const float* lds_dst,
                                                 const float* gsrc,
                                                 unsigned n) {
  uint64_t ga = (uint64_t)(uintptr_t)gsrc;
  u32x4 g0;
  g0[0] = 1u;                                            // count=1 (user D#)
  g0[1] = (uint32_t)(uintptr_t)lds_dst;                  // lds_addr
  g0[2] = (uint32_t)(ga & 0xffffffffu);                  // global_addr[31:0]
  g0[3] = (uint32_t)((ga >> 32) & 0x1ffffffu) | (2u << 30); // addr[56:32]|type=2
  i32x8 g1;
  g1[0] = (int)(2u << 16);                               // data_size=2 -> 4B
  g1[1] = (int)((n & 0xffffu) << 16);                    // tensor_dim0 lo16
  g1[2] = (int)(((n >> 16) & 0xffffu) | (1u << 16));     // dim0 hi16 | dim1=1
  g1[3] = (int)((n & 0xffffu) << 16);                    // tile_dim0
  g1[4] = 1;                                             // tile_dim1=1
  g1[5] = (int)n;                                        // dim0_stride lo32
  g1[6] = 0;
  g1[7] = 0;
  i32x4 z4 = {0, 0, 0, 0};
#if defined(__clang_major__) && (__clang_major__ >= 23)
  i32x8 z8 = {0, 0, 0, 0, 0, 0, 0, 0};
  __builtin_amdgcn_tensor_load_to_lds(g0, g1, z4, z4, z8, 0);
#else
  __builtin_amdgcn_tensor_load_to_lds(g0, g1, z4, z4, 0);
#endif
}

// One 16x16 output tile, K-loop in steps of 4. Works on LDS or global ptrs.
// A: M x K row-major (lda).  B: transB ? (N x K, ldb)  :  (K x N, ldb).
// C: transC ? C[col*ldc+row] : C[row*ldc+col].
__device__ __forceinline__ void gemm_tile(
    const float* A, int lda, const float* B, int ldb, int transB,
    float* C, int ldc, int transC, int m0, int n0, int K,
    const float* bias, int relu, float scale)
{
  int lane = threadIdx.x & 31;
  int half = lane >> 4;      // 0: K 0,1   1: K 2,3
  int l15  = lane & 15;
  int arow = m0 + l15;
  int bcol = n0 + l15;
  v8f acc = {0.f,0.f,0.f,0.f,0.f,0.f,0.f,0.f};
  for (int k = 0; k < K; k += 4) {
    int kb = k + 2 * half;
    v2f a, b;
    a.x = A[arow * lda + kb];
    a.y = A[arow * lda + kb + 1];
    if (transB) { b.x = B[bcol * ldb + kb];     b.y = B[bcol * ldb + kb + 1]; }
    else        { b.x = B[kb * ldb + bcol];     b.y = B[(kb + 1) * ldb + bcol]; }
    acc = wmma4(a, b, acc);
  }
  float bs = bias ? bias[bcol] : 0.0f;
  int rbase = m0 + half * 8;
#pragma unroll
  for (int v = 0; v < 8; ++v) {
    float xv = acc[v] * scale + bs;
    if (relu) xv = fmaxf(xv, 0.0f);
    if (transC) C[bcol * ldc + (rbase + v)] = xv;
    else        C[(rbase + v) * ldc + bcol] = xv;
  }
}

// All-wave tiled GEMM over an (M/16)x(N/16) grid of tiles. Wave-uniform loop
// so EXEC stays all-ones around the WMMA ops.
__device__ __forceinline__ void gemm_all(
    const float* A, int lda, const float* B, int ldb, int transB,
    float* C, int ldc, int transC, int M, int N, int K,
    const float* bias, int relu, float scale)
{
  int wave   = (int)(threadIdx.x >> 5);
  int nwaves = (int)(blockDim.x >> 5);
  int nt = N >> 4, mt = M >> 4;
  for (int t = wave; t < mt * nt; t += nwaves) {
    gemm_tile(A, lda, B, ldb, transB, C, ldc, transC,
              (t / nt) << 4, (t % nt) << 4, K, bias, relu, scale);
  }
}

__device__ __forceinline__ float wave_sum(float v) {
#pragma unroll
  for (int m = 16; m >= 1; m >>= 1) v += __shfl_xor(v, m, 32);
  return v;
}
__device__ __forceinline__ float wave_max(float v) {
#pragma unroll
  for (int m = 16; m >= 1; m >>= 1) v = fmaxf(v, __shfl_xor(v, m, 32));
  return v;
}

#define N_SAMPLES 32768
#define MODEL 64
#define NFRM 128
#define KSIZE 128

// ---------------------------------------------------------------------------
// Kernel 0: zero output accumulator
// ---------------------------------------------------------------------------
__global__ void zero_kernel(float* p, int n) {
  int i = blockIdx.x * blockDim.x + threadIdx.x;
  if (i < n) p[i] = 0.0f;
}

// ---------------------------------------------------------------------------
// Kernel 1: filterbank conv + pooling, refactored as prefix-sum + WMMA GEMM.
// One workgroup per batch. x[b] (128KB) staged into LDS by the Tensor Data
// Mover (one descriptor, issued by wave 0, TENSORcnt-tracked).
// spec[b,band,f] = (1/512) * sum_k filt[band,k] * W[f,k]
// W[f,k] = sum_{i in win_f} x[i + k - 64],  win_f = [f*256-256, f*256+256)∩[0,N)
// ---------------------------------------------------------------------------
__global__ void spec_kernel(const float* __restrict__ x,
                            const float* __restrict__ filt,
                            float* __restrict__ spec)
{
  extern __shared__ float sm[];
  float* X  = sm;              // 32768  : x[b], then in-place inclusive prefix
  float* W  = sm + 32768;      // 16384  : W[f][k]
  float* Bt = sm + 49152;      // 8192   : filt^T [k][band]
  float* S  = sm + 57344;      // 256    : per-thread chunk totals
  float* S2 = sm + 57600;      // 256    : exclusive scan of S
  const int b = blockIdx.x, tid = threadIdx.x;

  // TDM: DMA the whole batch row into LDS, no VGPR staging.
  if (tid < 32) {
    tdm_load_row_f32(X, x + b * N_SAMPLES, N_SAMPLES);
    __builtin_amdgcn_s_wait_tensorcnt(0);
  }
  __syncthreads();
  // local inclusive prefix over 128-chunk
  {
    float run = 0.0f;
    int base = tid * 128;
    for (int j = 0; j < 128; ++j) { run += X[base + j]; X[base + j] = run; }
    S[tid] = run;
  }
  __syncthreads();
  if (tid == 0) {
    float run = 0.0f;
    for (int t = 0; t < 256; ++t) { S2[t] = run; run += S[t]; }
  }
  __syncthreads();
  {
    float off = S2[tid];
    int base = tid * 128;
    for (int j = 0; j < 128; ++j) X[base + j] += off;
  }
  __syncthreads();
  // W[f][k] from clamped prefix differences
  for (int i = tid; i < NFRM * KSIZE; i += 256) {
    int f = i >> 7, k = i & 127;
    int lo = f * 256 - 256; if (lo < 0) lo = 0;
    int hi = f * 256 + 256;                      // <= 32768 always
    int a_ = lo - 64 + k - 1;                    // inclusive-prefix indices
    int b_ = hi - 64 + k - 1;
    float Qa = (a_ < 0) ? 0.0f : X[a_ < (N_SAMPLES - 1) ? a_ : (N_SAMPLES - 1)];
    float Qb = (b_ < 0) ? 0.0f : X[b_ < (N_SAMPLES - 1) ? b_ : (N_SAMPLES - 1)];
    W[i] = Qb - Qa;
  }
  // filt (64,1,128) -> Bt[k][band]
  for (int i = tid; i < KSIZE * 64; i += 256) {
    int k = i >> 6, n = i & 63;
    Bt[i] = filt[n * KSIZE + k];
  }
  __syncthreads();
  // GEMM: M=128 frames, N=64 bands, K=128; store transposed -> spec[b][band][f]
  gemm_all(W, KSIZE, Bt, 64, /*transB=*/0,
           spec + b * 64 * NFRM, NFRM, /*transC=*/1,
           NFRM, 64, KSIZE, nullptr, 0, 1.0f / 512.0f);
}

// ---------------------------------------------------------------------------
// Kernel 2: positional features + embed GEMM. One WG per batch.
// feat rows: [pos(1); sin(pos*2^j*pi) j=0..15; cos(...) (16); spec(64)] = 97,
// K padded to 100. h[b][t][c] = sum_k feat[k][t]*embed_w[c][k] + embed_b[c]
// ---------------------------------------------------------------------------
__global__ void embed_kernel(const float* __restrict__ spec,
                             const float* __restrict__ ew,
                             const float* __restrict__ eb,
                             float* __restrict__ h)
{
  extern __shared__ float sm[];
  float* A  = sm;            // 128 x 100 (feat^T, zero-padded K)
  float* Bt = sm + 12800;    // 100 x 64  (embed_w^T, zero-padded K)
  const int b = blockIdx.x, tid = threadIdx.x;
  const float PI = 3.14159265358979323846f;

  for (int i = tid; i < NFRM * 100; i += 256) {
    int t = i / 100, k = i - t * 100;
    float pos = (2.0f * (float)t) / 127.0f - 1.0f;
    float v;
    if (k == 0)       v = pos;
    else if (k < 17)  v = __sinf(pos * (float)(1 << (k - 1)) * PI);
    else if (k < 33)  v = __cosf(pos * (float)(1 << (k - 17)) * PI);
    else if (k < 97)  v = spec[b * 8192 + (k - 33) * NFRM + t];
    else              v = 0.0f;
    A[i] = v;
  }
  for (int i = tid; i < 100 * 64; i += 256) {
    int k = i >> 6, n = i & 63;
    Bt[i] = (k < 97) ? ew[n * 97 + k] : 0.0f;
  }
  __syncthreads();
  gemm_all(A, 100, Bt, 64, /*transB=*/0,
           h + b * NFRM * MODEL, MODEL, /*transC=*/0,
           NFRM, MODEL, 100, eb, 0, 1.0f);
}

// ---------------------------------------------------------------------------
// Transformer: one WG per batch, 4 layers fully in LDS (256KB), WMMA GEMMs.
// ---------------------------------------------------------------------------
__device__ __forceinline__ void ln_residual(float* Xs, const float* Rs,
                                            const float* g, const float* bt,
                                            int tid)
{
  int wave = tid >> 5, lane = tid & 31;
  for (int r = wave; r < NFRM; r += 8) {
    float x0 = Xs[r * MODEL + lane]      + Rs[r * MODEL + lane];
    float x1 = Xs[r * MODEL + lane + 32] + Rs[r * MODEL + lane + 32];
    float m  = wave_sum(x0 + x1) * (1.0f / 64.0f);
    float d0 = x0 - m, d1 = x1 - m;
    float vv = wave_sum(d0 * d0 + d1 * d1) * (1.0f / 64.0f);
    float inv = rsqrtf(vv + 1e-5f);
    Xs[r * MODEL + lane]      = d0 * inv * g[lane]      + bt[lane];
    Xs[r * MODEL + lane + 32] = d1 * inv * g[lane + 32] + bt[lane + 32];
  }
}

__global__ void xformer_kernel(const float* __restrict__ h_in,
                               const float* __restrict__ spec,
                               const float* __restrict__ qkv_w,
                               const float* __restrict__ qkv_b,
                               const float* __restrict__ ao_w,
                               const float* __restrict__ ao_b,
                               const float* __restrict__ ln1g,
                               const float* __restrict__ ln1b,
                               const float* __restrict__ f1w,
                               const float* __restrict__ f1b,
                               const float* __restrict__ f2w,
                               const float* __restrict__ f2b,
                               const float* __restrict__ ln2g,
                               const float* __restrict__ ln2b,
                               float* __restrict__ xf)
{
  extern __shared__ float sm[];
  float* Hs  = sm;                 // 128 x 64
  float* QKV = sm + 8192;          // 128 x 192
  float* SC  = sm + 32768;         // 128 x 128
  float* O   = sm + 49152;         // 128 x 64
  float* T1  = sm + 57344;         // 128 x 64
  const int b = blockIdx.x, tid = threadIdx.x;
  const int wave = tid >> 5, lane = tid & 31;

  // async global->LDS copy of the per-batch hidden state (16B aligned)
  for (int i = tid * 4; i < NFRM * MODEL; i += 1024)
    async_b128(Hs + i, h_in + b * 8192 + i);
  wait_async0();
  __syncthreads();

  for (int l = 0; l < 4; ++l) {
    // QKV projection
    gemm_all(Hs, MODEL, qkv_w + l * 192 * MODEL, MODEL, 1,
             QKV, 192, 0, NFRM, 192, MODEL, qkv_b + l * 192, 0, 1.0f);
    __syncthreads();
    for (int hd = 0; hd < 4; ++hd) {
      // scores = Q K^T / sqrt(16)
      gemm_all(QKV + hd * 16, 192, QKV + 64 + hd * 16, 192, 1,
               SC, NFRM, 0, NFRM, NFRM, 16, nullptr, 0, 0.25f);
      __syncthreads();
      // row softmax (wave per row)
      for (int r = wave; r < NFRM; r += 8) {
        float v[4], mx = -3.0e38f;
#pragma unroll
        for (int j = 0; j < 4; ++j) { v[j] = SC[r * NFRM + lane + 32 * j]; mx = fmaxf(mx, v[j]); }
        mx = wave_max(mx);
        float s = 0.0f;
#pragma unroll
        for (int j = 0; j < 4; ++j) { v[j] = __expf(v[j] - mx); s += v[j]; }
        s = wave_sum(s);
        float inv = 1.0f / s;
#pragma unroll
        for (int j = 0; j < 4; ++j) SC[r * NFRM + lane + 32 * j] = v[j] * inv;
      }
      __syncthreads();
      // O_head = softmax(scores) @ V
      gemm_all(SC, NFRM, QKV + 128 + hd * 16, 192, 0,
               O + hd * 16, MODEL, 0, NFRM, 16, NFRM, nullptr, 0, 1.0f);
      __syncthreads();
    }
    // attention out projection -> T1
    gemm_all(O, MODEL, ao_w + l * MODEL * MODEL, MODEL, 1,
             T1, MODEL, 0, NFRM, MODEL, MODEL, ao_b + l * MODEL, 0, 1.0f);
    __syncthreads();
    ln_residual(Hs, T1, ln1g + l * MODEL, ln1b + l * MODEL, tid);
    __syncthreads();
    // FFN
    gemm_all(Hs, MODEL, f1w + l * MODEL * MODEL, MODEL, 1,
             T1, MODEL, 0, NFRM, MODEL, MODEL, f1b + l * MODEL, 1, 1.0f);
    __syncthreads();
    gemm_all(T1, MODEL, f2w + l * MODEL * MODEL, MODEL, 1,
             O, MODEL, 0, NFRM, MODEL, MODEL, f2b + l * MODEL, 0, 1.0f);
    __syncthreads();
    ln_residual(Hs, O, ln2g + l * MODEL, ln2b + l * MODEL, tid);
    __syncthreads();
  }
  // xf[b][c][t] = h[t][c] + spec[b][c][t]
  for (int i = tid; i < 8192; i += 256) {
    int c = i >> 7, t = i & 127;
    xf[b * 8192 + i] = Hs[t * MODEL + c] + spec[b * 8192 + i];
  }
}

// ---------------------------------------------------------------------------
// Env softmax + top-16 + latent gather. One 128-thread WG per batch.
// ---------------------------------------------------------------------------
__global__ void env_topk_kernel(const float* __restrict__ xf,
                                const float* __restrict__ wenv,
                                const float* __restrict__ benv,
                                float* __restrict__ vals,
                                int* __restrict__ idxs,
                                float* __restrict__ lat)
{
  __shared__ float P[128], RV[128];
  __shared__ int   RI[128], selIdx[16];
  __shared__ float selVal[16];
  const int b = blockIdx.x, t = threadIdx.x;
  const float* xb = xf + b * 8192;

  float acc = benv[0];
  for (int c = 0; c < 64; ++c) acc += wenv[c] * xb[c * 128 + t];
  RV[t] = acc; __syncthreads();
  for (int s = 64; s; s >>= 1) { if (t < s) RV[t] = fmaxf(RV[t], RV[t + s]); __syncthreads(); }
  float mx = RV[0]; __syncthreads();
  float e = __expf(acc - mx);
  RV[t] = e; __syncthreads();
  for (int s = 64; s; s >>= 1) { if (t < s) RV[t] += RV[t + s]; __syncthreads(); }
  float denom = RV[0]; __syncthreads();
  P[t] = e / denom; __syncthreads();

  for (int a = 0; a < 16; ++a) {
    RV[t] = P[t]; RI[t] = t; __syncthreads();
    for (int s = 64; s; s >>= 1) {
      if (t < s) {
        float v2 = RV[t + s]; int i2 = RI[t + s];
        if (v2 > RV[t] || (v2 == RV[t] && i2 < RI[t])) { RV[t] = v2; RI[t] = i2; }
      }
      __syncthreads();
    }
    if (t == 0) { selIdx[a] = RI[0]; selVal[a] = RV[0]; P[RI[0]] = -1.0f; }
    __syncthreads();
  }
  if (t < 16) { vals[b * 16 + t] = selVal[t]; idxs[b * 16 + t] = selIdx[t]; }
  for (int i = t; i < 16 * 64; i += 128) {
    int a = i >> 6, c = i & 63;
    lat[(b * 16 + a) * 64 + c] = xb[c * 128 + selIdx[a]];
  }
}

// ---------------------------------------------------------------------------
// Up-projection GEMM: t0(256x512) = latents(256x64) @ up_w^T + up_b. WMMA,
// one 16x16 tile per wave, 64 blocks x 8 waves = 512 tiles.
// ---------------------------------------------------------------------------
__global__ void up_kernel(const float* __restrict__ lat,
                          const float* __restrict__ upw,
                          const float* __restrict__ upb,
                          float* __restrict__ t0)
{
  int tileg = blockIdx.x * 8 + (threadIdx.x >> 5); // 0..511
  int nt = 512 / 16;                               // 32 n-tiles
  int m0 = (tileg / nt) << 4, n0 = (tileg % nt) << 4;
  gemm_tile(lat, 64, upw, 64, /*transB=*/1, t0, 512, /*transC=*/0,
            m0, n0, 64, upb, 0, 1.0f);
}

// ---------------------------------------------------------------------------
// Per-atom deconv pyramid, entirely in LDS (300KB: R0 128K | R1 40K | R2 128K)
// conv_t: y[o][t] = b[o] + sum_i sum_{k=k0,k0+4} w[i][o][7-k] * x[i][(t+k-5)/4]
//   with k0 = (5 - t) mod 4; layers 4/5 chunked 8x4096 so in+out fit the pool.
// Final: max-normalize, scale by top-k value, atomicAdd scatter into output.
// ---------------------------------------------------------------------------
__device__ __forceinline__ void deconv_layer(const float* __restrict__ in,
                                             float* __restrict__ out,
                                             int Cin, int Cout, int Lin,
                                             const float* __restrict__ w,
                                             const float* __restrict__ bias,
                                             int lrelu)
{
  int Lout = 4 * Lin;
  int total = Cout * Lout;
  for (int i = threadIdx.x; i < total; i += 256) {
    int o = i / Lout, t = i - o * Lout;
    float acc = bias[o];
    int k0 = (5 - (t & 3)) & 3;
    for (int k = k0; k < 8; k += 4) {
      int m = (t + k - 5) >> 2;
      if (m < 0 || m >= Lin) continue;
      for (int ci = 0; ci < Cin; ++ci)
        acc += w[(ci * Cout + o) * 8 + (7 - k)] * in[ci * Lin + m];
    }
    if (lrelu && acc < 0.0f) acc *= 0.2f;
    out[i] = acc;
  }
}

__global__ void atoms_kernel(const float* __restrict__ t0,
                             const float* __restrict__ tw0, const float* __restrict__ tb0,
                             const float* __restrict__ tw1, const float* __restrict__ tb1,
                             const float* __restrict__ tw2, const float* __restrict__ tb2,
                             const float* __restrict__ tw3, const float* __restrict__ tb3,
                             const float* __restrict__ tw4, const float* __restrict__ tb4,
                             const float* __restrict__ tw5, const float* __restrict__ tb5,
                             const float* __restrict__ vals,
                             const int* __restrict__ idxs,
                             float* __restrict__ out)
{
  extern __shared__ float sm[];
  float* R0 = sm;            // 32768 floats
  float* R1 = sm + 32768;    // 10240 floats
  float* R2 = sm + 43008;    // 32768 floats
  const int a = blockIdx.x, b = a >> 4, tid = threadIdx.x;

  // hint: pull shared deconv weights toward the caches
  if (tid < 64) __builtin_prefetch(tw2 + tid * 256, 0, 1);

  // async global->LDS copy of this atom's seed row (512 floats, 16B aligned)
  if (tid < 128) async_b128(R1 + tid * 4, t0 + a * 512 + tid * 4);
  wait_async0();
  __syncthreads();
  deconv_layer(R1, R2, 64, 64, 8,   tw0, tb0, 1); __syncthreads(); // 64x32
  deconv_layer(R2, R1, 64, 64, 32,  tw1, tb1, 1); __syncthreads(); // 64x128
  deconv_layer(R1, R2, 64, 32, 128, tw2, tb2, 1); __syncthreads(); // 32x512
  deconv_layer(R2, R0, 32, 16, 512, tw3, tb3, 1); __syncthreads(); // 16x2048 in R0

  // layers 4+5, chunked over final time (8 chunks of 4096)
  for (int c = 0; c < 8; ++c) {
    int T0 = c * 4096;
    int m_lo = T0 / 4 - 2; if (m_lo < 0) m_lo = 0;
    int m_hi = T0 / 4 + 1024; if (m_hi > 8191) m_hi = 8191;
    int cnt = m_hi - m_lo + 1;                 // <= 1027
    // layer4 chunk -> R1[o*1027 + (m - m_lo)]
    for (int i = tid; i < 8 * cnt; i += 256) {
      int o = i / cnt, j = i - o * cnt, m = m_lo + j;
      float acc = tb4[o];
      int k0 = (5 - (m & 3)) & 3;
      for (int k = k0; k < 8; k += 4) {
        int mm = (m + k - 5) >> 2;
        if (mm < 0 || mm >= 2048) continue;
        for (int ci = 0; ci < 16; ++ci)
          acc += tw4[(ci * 8 + o) * 8 + (7 - k)] * R0[ci * 2048 + mm];
      }
      if (acc < 0.0f) acc *= 0.2f;
      R1[o * 1027 + j] = acc;
    }
    __syncthreads();
    // layer5 chunk -> R2[T0 + tl]
    for (int tl = tid; tl < 4096; tl += 256) {
      int t = T0 + tl;
      float acc = tb5[0];
      int k0 = (5 - (t & 3)) & 3;
      for (int k = k0; k < 8; k += 4) {
        int m = (t + k - 5) >> 2;
        if (m < 0 || m >= 8192) continue;
        int j = m - m_lo;
        for (int ci = 0; ci < 8; ++ci)
          acc += tw5[ci * 8 + (7 - k)] * R1[ci * 1027 + j];
      }
      R2[t] = acc;
    }
    __syncthreads();
  }
  // max over the 32768-sample atom
  {
    float mx = -3.0e38f;
    for (int i = tid; i < N_SAMPLES; i += 256) mx = fmaxf(mx, R2[i]);
    R1[tid] = mx; __syncthreads();
    for (int s = 128; s; s >>= 1) { if (tid < s) R1[tid] = fmaxf(R1[tid], R1[tid + s]); __syncthreads(); }
  }
  float scale = vals[a] / (R1[0] + 1e-8f);
  int pos0 = idxs[a] * 256;
  for (int tl = tid; tl < N_SAMPLES; tl += 256) {
    int pos = pos0 + tl;
    if (pos < N_SAMPLES)
      atomicAdd(&out[b * N_SAMPLES + pos], R2[tl] * scale);
  }
}

// ---------------------------------------------------------------------------
extern "C" void kernel_launch(void* const* d_in, const int* in_sizes, int n_in,
                              void* d_out, int out_size, void* d_ws, size_t ws_size,
                              hipStream_t stream) {
  const float* x     = (const float*)d_in[0];
  const float* fbf   = (const float*)d_in[1];
  const float* ew    = (const float*)d_in[2];
  const float* ebv   = (const float*)d_in[3];
  const float* qkvw  = (const float*)d_in[4];
  const float* qkvb  = (const float*)d_in[5];
  const float* aow   = (const float*)d_in[6];
  const float* aob   = (const float*)d_in[7];
  const float* ln1g  = (const float*)d_in[8];
  const float* ln1b  = (const float*)d_in[9];
  const float* f1w   = (const float*)d_in[10];
  const float* f1b   = (const float*)d_in[11];
  const float* f2w   = (const float*)d_in[12];
  const float* f2b   = (const float*)d_in[13];
  const float* ln2g  = (const float*)d_in[14];
  const float* ln2b  = (const float*)d_in[15];
  const float* tew   = (const float*)d_in[16];
  const float* teb   = (const float*)d_in[17];
  const float* upw   = (const float*)d_in[18];
  const float* upb   = (const float*)d_in[19];
  const float* tw0   = (const float*)d_in[20];
  const float* tb0   = (const float*)d_in[21];
  const float* tw1   = (const float*)d_in[22];
  const float* tb1   = (const float*)d_in[23];
  const float* tw2   = (const float*)d_in[24];
  const float* tb2   = (const float*)d_in[25];
  const float* tw3   = (const float*)d_in[26];
  const float* tb3   = (const float*)d_in[27];
  const float* tw4   = (const float*)d_in[28];
  const float* tb4   = (const float*)d_in[29];
  const float* tw5   = (const float*)d_in[30];
  const float* tb5   = (const float*)d_in[31];
  float* out = (float*)d_out;

  float* ws   = (float*)d_ws;
  float* spec = ws;                 // 16*64*128
  float* h    = ws + 131072;        // 16*128*64
  float* xf   = ws + 262144;        // 16*64*128
  float* lat  = ws + 393216;        // 256*64
  float* t0   = ws + 409600;        // 256*512
  float* vals = ws + 540672;        // 256
  int*   idxs = (int*)(ws + 540928);// 256

  zero_kernel<<<2048, 256, 0, stream>>>(out, 16 * N_SAMPLES);
  spec_kernel<<<16, 256, 57856 * sizeof(float), stream>>>(x, fbf, spec);
  embed_kernel<<<16, 256, 19200 * sizeof(float), stream>>>(spec, ew, ebv, h);
  xformer_kernel<<<16, 256, 65536 * sizeof(float), stream>>>(
      h, spec, qkvw, qkvb, aow, aob, ln1g, ln1b,
      f1w, f1b, f2w, f2b, ln2g, ln2b, xf);
  env_topk_kernel<<<16, 128, 0, stream>>>(xf, tew, teb, vals, idxs, lat);
  up_kernel<<<64, 256, 0, stream>>>(lat, upw, upb, t0);
  atoms_kernel<<<256, 256, 75776 * sizeof(float), stream>>>(
      t0, tw0, tb0, tw1, tb1, tw2, tb2, tw3, tb3, tw4, tb4, tw5, tb5,
      vals, idxs, out);
}